// Network_56349970923535
// MI455X (gfx1250) — compile-verified
//
#include <hip/hip_runtime.h>
#include <hip/hip_bf16.h>

typedef float v4f __attribute__((ext_vector_type(4)));
typedef float v2f __attribute__((ext_vector_type(2)));
typedef int   v2i __attribute__((ext_vector_type(2)));

// Flip to 0 if the async-to-LDS inline asm is rejected by the assembler.
#ifndef USE_ASYNC_LDS
#define USE_ASYNC_LDS 1
#endif

namespace {
constexpr int kC = 512, kH = 50, kW = 75, kN = 512;
constexpr int kPool = 7, kPre = 14;
constexpr int kCT = 4;                       // channels per block (b128 gather)
constexpr int kCG = kC / kCT;                // 128 channel groups
constexpr int kRSplit = 4;                   // ROI splits -> 512 blocks total
constexpr int kRoisPerBlk = kN / kRSplit;    // 128
constexpr int kChunk = 8;                    // ROIs per coefficient chunk
constexpr int kNChunk = kRoisPerBlk / kChunk;// 16
constexpr int kPlane = kH * kW;              // 3750 cells
constexpr int kThreads = 256;                // 8 wave32
constexpr unsigned kPlaneBytes = kPlane * 16;          // 60000 B, [cell][c4]
constexpr unsigned kCoefBytes  = kChunk * kPre * 8;    // 896 B per table
constexpr unsigned kSmemBytes  = kPlaneBytes + 4 * kCoefBytes; // 63584 B
}

__device__ __forceinline__ v4f vmax4(v4f a, v4f b) {
  v4f r;
  r.x = __builtin_fmaxf(a.x, b.x);
  r.y = __builtin_fmaxf(a.y, b.y);
  r.z = __builtin_fmaxf(a.z, b.z);
  r.w = __builtin_fmaxf(a.w, b.w);
  return r;
}

__global__ __launch_bounds__(kThreads) void roi_align_max_kernel(
    const float* __restrict__ bottom, const float* __restrict__ rois,
    float* __restrict__ out) {
  extern __shared__ __align__(16) char smem[];
  v4f* s_plane = reinterpret_cast<v4f*>(smem);
  v2i* s_xi = reinterpret_cast<v2i*>(smem + kPlaneBytes);
  v2f* s_xw = reinterpret_cast<v2f*>(smem + kPlaneBytes + 1 * kCoefBytes);
  v2i* s_yi = reinterpret_cast<v2i*>(smem + kPlaneBytes + 2 * kCoefBytes);
  v2f* s_yw = reinterpret_cast<v2f*>(smem + kPlaneBytes + 3 * kCoefBytes);

  const int tid = threadIdx.x;
  const int cg  = blockIdx.x;   // channel group (4 channels)
  const int rsp = blockIdx.y;   // ROI split

  // ---- Stage 4 channel planes into LDS, interleaved as [cell][c0..c3] ----
  // Coalesced global reads; per-lane scattered LDS writes (stride 16B).
  // Outer channel loop (unrolled) avoids per-element div/mod address math.
  const float* src = bottom + (size_t)cg * kCT * kPlane;
  float* planef = reinterpret_cast<float*>(s_plane);
#pragma unroll
  for (int c = 0; c < kCT; ++c) {
    const float* srcc = src + (size_t)c * kPlane;
    float* dstc = planef + c;
    for (int cell = tid; cell < kPlane; cell += kThreads) {
#if USE_ASYNC_LDS
      // CDNA5 async copy: copies 4 bytes from the global address in vaddr to the LDS address in vdst; tracked on ASYNCcnt.
      unsigned lds_off = (unsigned)(size_t)(dstc + cell * kCT);
      asm volatile("global_load_async_to_lds_b32 %0, %1, off"
                   :: "v"(lds_off), "v"(srcc + cell)
                   : "memory");
#else
      dstc[cell * kCT] = srcc[cell];
#endif
    }
  }
#if USE_ASYNC_LDS
  asm volatile("s_wait_asynccnt 0x0" ::: "memory");
#endif
  __syncthreads();

  const size_t out_cbase = (size_t)cg * kCT * (kPool * kPool);

  for (int chunk = 0; chunk < kNChunk; ++chunk) {
    const int roi0 = rsp * kRoisPerBlk + chunk * kChunk;

    // ---- Per-ROI separable sampling coefficients (mirrors reference) ----
    if (tid < kChunk * 2 * kPre) {          // 8 ROIs * (14 x + 14 y) = 224
      int r = tid / (2 * kPre);
      int k = tid - r * 2 * kPre;
      bool isY = (k >= kPre);
      int idx = isY ? (k - kPre) : k;
      const float* rp = rois + (size_t)(roi0 + r) * 5;
      float c1  = (isY ? rp[2] : rp[1]) * (1.0f / 16.0f);
      float c2  = (isY ? rp[4] : rp[3]) * (1.0f / 16.0f);
      float dm1 = isY ? (float)(kH - 1) : (float)(kW - 1);
      float s   = (c2 - c1) / dm1;
      float tq  = (c1 + c2 - dm1) / dm1;
      float b   = -1.0f + (float)idx * (2.0f / 13.0f);   // linspace(-1,1,14)
      float g   = s * b + tq;
      float pos = (g + 1.0f) * 0.5f * dm1;
      float p0  = floorf(pos);
      float w1  = pos - p0;
      float w0  = 1.0f - w1;
      float p1  = p0 + 1.0f;
      float w0m = (p0 >= 0.0f && p0 <= dm1) ? w0 : 0.0f; // zero-pad mask
      float w1m = (p1 >= 0.0f && p1 <= dm1) ? w1 : 0.0f;
      int i0 = (int)fminf(fmaxf(p0, 0.0f), dm1);         // clip
      int i1 = (int)fminf(fmaxf(p1, 0.0f), dm1);
      int scale = isY ? kW : 1;                          // row index in cells
      v2i iv; iv.x = i0 * scale; iv.y = i1 * scale;
      v2f wv; wv.x = w0m; wv.y = w1m;
      int slot = r * kPre + idx;
      if (isY) { s_yi[slot] = iv; s_yw[slot] = wv; }
      else     { s_xi[slot] = iv; s_xw[slot] = wv; }
    }
    __syncthreads();

    // ---- Sample + 2x2 max-pool: 4 channels per lane via ds_load_b128 ----
    for (int t = tid; t < kChunk * kPool * kPool; t += kThreads) {
      int r    = t / (kPool * kPool);
      int cell = t - r * (kPool * kPool);
      int py   = cell / kPool;
      int px   = cell - py * kPool;
      int xs = r * kPre + px * 2;
      int ys = r * kPre + py * 2;
      v2i xb0 = s_xi[xs],     xb1 = s_xi[xs + 1];
      v2f xw0 = s_xw[xs],     xw1 = s_xw[xs + 1];
      v2i yb0 = s_yi[ys],     yb1 = s_yi[ys + 1];
      v2f yw0 = s_yw[ys],     yw1 = s_yw[ys + 1];

      auto bil = [&](v2i yb, v2f wy, v2i xb, v2f wx) -> v4f {
        v4f v00 = s_plane[yb.x + xb.x];
        v4f v01 = s_plane[yb.x + xb.y];
        v4f v10 = s_plane[yb.y + xb.x];
        v4f v11 = s_plane[yb.y + xb.y];
        return (v00 * wx.x + v01 * wx.y) * wy.x +
               (v10 * wx.x + v11 * wx.y) * wy.y;
      };

      v4f acc = bil(yb0, yw0, xb0, xw0);
      acc = vmax4(acc, bil(yb0, yw0, xb1, xw1));
      acc = vmax4(acc, bil(yb1, yw1, xb0, xw0));
      acc = vmax4(acc, bil(yb1, yw1, xb1, xw1));

      int n = roi0 + r;
      float* op = out + (size_t)n * (kC * kPool * kPool) + out_cbase + cell;
      op[0 * kPool * kPool] = acc.x;
      op[1 * kPool * kPool] = acc.y;
      op[2 * kPool * kPool] = acc.z;
      op[3 * kPool * kPool] = acc.w;
    }
    __syncthreads();  // protect coefficient tables before next chunk
  }
}

extern "C" void kernel_launch(void* const* d_in, const int* in_sizes, int n_in,
                              void* d_out, int out_size, void* d_ws, size_t ws_size,
                              hipStream_t stream) {
  (void)in_sizes; (void)n_in; (void)d_ws; (void)ws_size; (void)out_size;
  const float* bottom = (const float*)d_in[0];  // [1,512,50,75] f32
  const float* rois   = (const float*)d_in[1];  // [512,5] f32
  float* out = (float*)d_out;                   // [512,512,7,7] f32
  dim3 grid(kCG, kRSplit);                      // 128 x 4 = 512 blocks
  roi_align_max_kernel<<<grid, kThreads, kSmemBytes, stream>>>(bottom, rois, out);
}